// GCN_18356690223924
// MI455X (gfx1250) — compile-verified
//
#include <hip/hip_runtime.h>
#include <hip/hip_bf16.h>

#define NNODES 20000
#define NEDGES 320000
#define MAXW   512
#define BN_EPS 1e-5f

typedef __attribute__((ext_vector_type(2))) float v2f;
typedef __attribute__((ext_vector_type(8))) float v8f;

// Device-pass-only sanity check: AMDGCN target builtins are invisible to the host pass.
#if defined(__HIP_DEVICE_COMPILE__) && !__has_builtin(__builtin_amdgcn_wmma_f32_16x16x4_f32)
#error "gfx1250 f32 WMMA builtin not available on device"
#endif

// ---------------- graph normalization ----------------
__global__ void k_deg_init(float* __restrict__ deg) {
    int i = blockIdx.x * blockDim.x + threadIdx.x;
    if (i < NNODES) deg[i] = 1.0f;               // self loop
}
__global__ void k_deg_acc(const int* __restrict__ dst, float* __restrict__ deg) {
    int e = blockIdx.x * blockDim.x + threadIdx.x;
    if (e < NEDGES) atomicAdd(&deg[dst[e]], 1.0f);
}
__global__ void k_dinv(float* __restrict__ deg) {
    int i = blockIdx.x * blockDim.x + threadIdx.x;
    if (i < NNODES) deg[i] = rsqrtf(deg[i]);     // deg >= 1 always
}
__global__ void k_norm(const int* __restrict__ src, const int* __restrict__ dst,
                       const float* __restrict__ dinv, float* __restrict__ nrm) {
    int e = blockIdx.x * blockDim.x + threadIdx.x;
    if (e < NEDGES) nrm[e] = dinv[src[e]] * dinv[dst[e]];
}

// ---------------- fp32 WMMA GEMM: out[N,dout] = A[N,din] @ W[din,dout] ----------------
// one wave per 16x16 output tile; K stepped by 4 via V_WMMA_F32_16X16X4_F32
__global__ __launch_bounds__(32)
void k_gemm_wmma(const float* __restrict__ A, const float* __restrict__ W,
                 float* __restrict__ out, int din, int dout, int ntiles) {
    int tile  = blockIdx.x;
    int mtile = tile / ntiles;
    int ntile = tile - mtile * ntiles;
    int lane  = threadIdx.x;                 // 0..31, wave32
    int row0  = mtile << 4;
    int col0  = ntile << 4;
    int m     = lane & 15;
    int koff  = (lane >> 4) << 1;            // lanes 0-15: K=0,1 ; lanes 16-31: K=2,3
    int col   = col0 + (lane & 15);
    float msk = (col < dout) ? 1.0f : 0.0f;  // mask w/o divergent loads (EXEC stays all-1s)
    int colc  = (col < dout) ? col : 0;

    v8f acc = {0.f, 0.f, 0.f, 0.f, 0.f, 0.f, 0.f, 0.f};
    const float* arow = A + (size_t)(row0 + m) * din;
    for (int k0 = 0; k0 < din; k0 += 4) {
        v2f a, b;
        a.x = arow[k0 + koff];
        a.y = arow[k0 + koff + 1];
        b.x = W[(size_t)(k0 + koff)     * dout + colc] * msk;
        b.y = W[(size_t)(k0 + koff + 1) * dout + colc] * msk;
        acc = __builtin_amdgcn_wmma_f32_16x16x4_f32(false, a, false, b,
                                                    (short)0, acc, false, false);
    }
    if (col < dout) {
        int rbase = row0 + ((lane >> 4) << 3);   // VGPR v: lanes0-15 -> M=v, lanes16-31 -> M=v+8
#pragma unroll
        for (int v = 0; v < 8; ++v)
            out[(size_t)(rbase + v) * dout + col] = acc[v];
    }
}

// ---------------- aggregation ----------------
__global__ void k_agg_init(const float* __restrict__ h, const float* __restrict__ dinv,
                           float* __restrict__ agg, int dout) {
    int idx = blockIdx.x * blockDim.x + threadIdx.x;
    if (idx < NNODES * dout) {
        int i = idx / dout;
        float d = dinv[i];
        agg[idx] = d * d * h[idx];               // self-loop contribution
    }
}
__global__ void k_scatter4(const int* __restrict__ src, const int* __restrict__ dst,
                           const float* __restrict__ nrm, const float* __restrict__ h,
                           float* __restrict__ agg, int dout4) {
    int idx = blockIdx.x * blockDim.x + threadIdx.x;
    if (idx >= NEDGES * dout4) return;
    int e  = idx / dout4;
    int f4 = (idx - e * dout4) << 2;
    int dout = dout4 << 2;
    int s = src[e], d = dst[e];
    float w = nrm[e];
    const float4 v = *(const float4*)(h + (size_t)s * dout + f4);
    float* ap = agg + (size_t)d * dout + f4;
    atomicAdd(ap + 0, w * v.x);
    atomicAdd(ap + 1, w * v.y);
    atomicAdd(ap + 2, w * v.z);
    atomicAdd(ap + 3, w * v.w);
}
__global__ void k_scatter1(const int* __restrict__ src, const int* __restrict__ dst,
                           const float* __restrict__ nrm, const float* __restrict__ h,
                           float* __restrict__ agg, int dout) {
    int idx = blockIdx.x * blockDim.x + threadIdx.x;
    if (idx >= NEDGES * dout) return;
    int e = idx / dout;
    int f = idx - e * dout;
    atomicAdd(&agg[(size_t)dst[e] * dout + f],
              nrm[e] * h[(size_t)src[e] * dout + f]);
}

// ---------------- bias + BN(eval) + ReLU ----------------
__global__ void k_bn_relu(const float* __restrict__ agg, const float* __restrict__ bias,
                          const float* __restrict__ gamma, const float* __restrict__ beta,
                          const float* __restrict__ mean, const float* __restrict__ var,
                          float* __restrict__ out, int dout) {
    int idx = blockIdx.x * blockDim.x + threadIdx.x;
    if (idx < NNODES * dout) {
        int f = idx % dout;
        float inv = rsqrtf(var[f] + BN_EPS);
        float v = (agg[idx] + bias[f] - mean[f]) * inv * gamma[f] + beta[f];
        out[idx] = v > 0.0f ? v : 0.0f;
    }
}

// ---------------- final bias + log_softmax (dout == 10) ----------------
__global__ void k_logsoftmax(const float* __restrict__ agg, const float* __restrict__ bias,
                             float* __restrict__ out) {
    int i = blockIdx.x * blockDim.x + threadIdx.x;
    if (i >= NNODES) return;
    float v[10];
    float mx = -1e30f;
#pragma unroll
    for (int c = 0; c < 10; ++c) { v[c] = agg[i * 10 + c] + bias[c]; mx = fmaxf(mx, v[c]); }
    float s = 0.0f;
#pragma unroll
    for (int c = 0; c < 10; ++c) s += expf(v[c] - mx);
    float lse = mx + logf(s);
#pragma unroll
    for (int c = 0; c < 10; ++c) out[i * 10 + c] = v[c] - lse;
}

extern "C" void kernel_launch(void* const* d_in, const int* in_sizes, int n_in,
                              void* d_out, int out_size, void* d_ws, size_t ws_size,
                              hipStream_t stream) {
    const float* x   = (const float*)d_in[0];
    const int* ei    = (const int*)d_in[1];
    const int* srcI  = ei;                  // edge_index[0]
    const int* dstI  = ei + NEDGES;         // edge_index[1]
    const float *W[7], *B[7];
    for (int i = 0; i < 7; ++i) { W[i] = (const float*)d_in[2 + 2*i]; B[i] = (const float*)d_in[3 + 2*i]; }
    const float *G[6], *Bt[6], *Mu[6], *Va[6];
    for (int i = 0; i < 6; ++i) {
        G[i]  = (const float*)d_in[16 + 4*i];
        Bt[i] = (const float*)d_in[17 + 4*i];
        Mu[i] = (const float*)d_in[18 + 4*i];
        Va[i] = (const float*)d_in[19 + 4*i];
    }

    float* ws   = (float*)d_ws;
    float* dinv = ws;                                   // N
    float* nrm  = dinv + NNODES;                        // E
    float* buf0 = nrm  + NEDGES;                        // N*512 (layer input, L>=1)
    float* bufP = buf0 + (size_t)NNODES * MAXW;         // N*512 (GEMM output)
    float* bufA = bufP + (size_t)NNODES * MAXW;         // N*512 (aggregation)

    auto nb = [](long n, int t) { return (unsigned)((n + t - 1) / t); };

    k_deg_init<<<nb(NNODES, 256), 256, 0, stream>>>(dinv);
    k_deg_acc <<<nb(NEDGES, 256), 256, 0, stream>>>(dstI, dinv);
    k_dinv    <<<nb(NNODES, 256), 256, 0, stream>>>(dinv);
    k_norm    <<<nb(NEDGES, 256), 256, 0, stream>>>(srcI, dstI, dinv, nrm);

    const int dims[8] = {128, 16, 32, 64, 128, 256, 512, 10};
    const float* hin = x;
    for (int L = 0; L < 7; ++L) {
        int din = dims[L], dout = dims[L + 1];
        int ntiles = (dout + 15) / 16;
        k_gemm_wmma<<<(NNODES / 16) * ntiles, 32, 0, stream>>>(hin, W[L], bufP, din, dout, ntiles);
        k_agg_init<<<nb((long)NNODES * dout, 256), 256, 0, stream>>>(bufP, dinv, bufA, dout);
        if ((dout & 3) == 0)
            k_scatter4<<<nb((long)NEDGES * (dout >> 2), 256), 256, 0, stream>>>(
                srcI, dstI, nrm, bufP, bufA, dout >> 2);
        else
            k_scatter1<<<nb((long)NEDGES * dout, 256), 256, 0, stream>>>(
                srcI, dstI, nrm, bufP, bufA, dout);
        if (L < 6) {
            k_bn_relu<<<nb((long)NNODES * dout, 256), 256, 0, stream>>>(
                bufA, B[L], G[L], Bt[L], Mu[L], Va[L], buf0, dout);
            hin = buf0;
        } else {
            k_logsoftmax<<<nb(NNODES, 256), 256, 0, stream>>>(bufA, B[L], (float*)d_out);
        }
    }
}